// Model11_64630667870280
// MI455X (gfx1250) — compile-verified
//
#include <hip/hip_runtime.h>

typedef __attribute__((ext_vector_type(2))) float v2f;
typedef __attribute__((ext_vector_type(8))) float v8f;

#define NODE_DIM 8
#define EDGE_DIM 4
#define IN_DIM   20   // 2*NODE_DIM + EDGE_DIM
#define MSG_DIM  10
#define HID_DIM  5

// ---------------------------------------------------------------------------
// Zero the accumulation workspace (node messages + graph pool). Must run every
// call: the later kernels accumulate with atomics.
// ---------------------------------------------------------------------------
__global__ void zero_ws(float* __restrict__ p, int n) {
    int i = blockIdx.x * blockDim.x + threadIdx.x;
    const int stride = gridDim.x * blockDim.x;
    for (; i < n; i += stride) p[i] = 0.0f;
}

// Concatenated per-edge input feature: [x_src(8) | x_dst(8) | e_attr(4)]
// All-int indexing (max offset ~13M elements) so the compiler can use the
// SGPR-base + 32-bit-voffset global addressing form.
__device__ __forceinline__ float edge_feature(const float* __restrict__ na,
                                              const float* __restrict__ ea,
                                              int s, int d, int e, int k) {
    if (k < NODE_DIM)     return na[s * NODE_DIM + k];
    if (k < 2 * NODE_DIM) return na[d * NODE_DIM + (k - NODE_DIM)];
    return ea[e * EDGE_DIM + (k - 2 * NODE_DIM)];
}

// ---------------------------------------------------------------------------
// Edge phase: per-wave 16-edge tiles, msg = relu([16x20] @ Wm[20x10] + bm)
// via 5 chained V_WMMA_F32_16X16X4_F32 (exact fp32, K = 5*4 = 20), then
// atomic scatter-sum into x[dst].
//
// A layout (32-bit A 16x4, ISA 7.12.2): lane L holds row M=L%16;
//   lanes 0-15: VGPR{0,1}=K{0,1}; lanes 16-31: VGPR{0,1}=K{2,3}.
// B layout (4x16, mirrored): lane L holds col N=L%16, same K-per-lane map.
// D layout (16x16 f32): lane L col N=L%16; VGPR r row M = r + 8*(L>=16).
//
// Main loop covers FULL tiles only (no per-element bounds checks, no exec
// churn in the scatter). Remainder edges (< 16) go through a rarely-taken
// scalar tail behind one uniform branch.
// ---------------------------------------------------------------------------
__global__ __launch_bounds__(256) void edge_mlp_wmma(
    const int*   __restrict__ src,
    const int*   __restrict__ dst,
    const float* __restrict__ node_attr,
    const float* __restrict__ edge_attr,
    const float* __restrict__ Wm,
    const float* __restrict__ bm,
    float*       __restrict__ x,      // [n_nodes, 10] accumulator
    int n_edges) {
    const int lane = threadIdx.x & 31;
    const int wave = threadIdx.x >> 5;
    const int col  = lane & 15;       // N index (matrix column / output channel)
    const int hi   = lane >> 4;       // 0: lanes 0-15, 1: lanes 16-31

    // --- tile-invariant B fragments + bias (masked select, no divergence) ---
    const int   colc = (col < MSG_DIM) ? col : 0;        // safe clamped column
    const float bsel = (col < MSG_DIM) ? 1.0f : 0.0f;    // zero pad cols 10..15
    v2f bfrag[5];
#pragma unroll
    for (int c = 0; c < 5; ++c) {
        const int k0 = c * 4 + hi * 2;                   // Wm is [20,10] row-major
        bfrag[c].x = Wm[k0 * MSG_DIM + colc] * bsel;
        bfrag[c].y = Wm[(k0 + 1) * MSG_DIM + colc] * bsel;
    }
    const float bias = bm[colc];

    const int n_full      = n_edges >> 4;                // full 16-edge tiles
    const int wave_stride = gridDim.x * 8;

    for (int tile = blockIdx.x * 8 + wave; tile < n_full;
         tile += wave_stride) {                          // wave-uniform loop
        const int e0 = tile << 4;
        const int eA = e0 + col;                         // always in-bounds
        const int sA = src[eA];
        const int dA = dst[eA];

        v8f acc = {};
#pragma unroll
        for (int c = 0; c < 5; ++c) {                    // K chunks of 4 -> K=20
            const int k0 = c * 4 + hi * 2;
            v2f a;
            a.x = edge_feature(node_attr, edge_attr, sA, dA, eA, k0);
            a.y = edge_feature(node_attr, edge_attr, sA, dA, eA, k0 + 1);
            acc = __builtin_amdgcn_wmma_f32_16x16x4_f32(
                /*neg_a=*/false, a, /*neg_b=*/false, bfrag[c],
                /*c_mod=*/(short)0, acc, /*reuse_a=*/false, /*reuse_b=*/false);
        }

        // Broadcast dst node indices for this lane's 8 output rows. Full EXEC
        // here (ds_bpermute reads from disabled lanes return zero), so do this
        // before the divergent scatter guard.
        int dnode[8];
#pragma unroll
        for (int r = 0; r < 8; ++r)
            dnode[r] = __shfl(dA, r + hi * 8, 32);       // lane m holds dst[e0+m]

        // Bias + ReLU + scatter-sum (full tile: no per-element guards).
        if (col < MSG_DIM) {
#pragma unroll
            for (int r = 0; r < 8; ++r) {
                float v = acc[r] + bias;
                v = v > 0.0f ? v : 0.0f;
                atomicAdd(&x[dnode[r] * MSG_DIM + col], v);
            }
        }
    }

    // ---- scalar tail for n_edges % 16 edges (uniform branch; usually 0) ----
    const int rem = n_edges - (n_full << 4);
    if (rem > 0) {
        const int gid = blockIdx.x * 256 + threadIdx.x;
        if (gid < rem * MSG_DIM) {
            const int e = (n_full << 4) + gid / MSG_DIM;
            const int j = gid % MSG_DIM;
            const int s = src[e];
            const int d = dst[e];
            float acc = bm[j];
#pragma unroll
            for (int k = 0; k < NODE_DIM; ++k)
                acc = fmaf(node_attr[s * NODE_DIM + k], Wm[k * MSG_DIM + j], acc);
#pragma unroll
            for (int k = 0; k < NODE_DIM; ++k)
                acc = fmaf(node_attr[d * NODE_DIM + k],
                           Wm[(NODE_DIM + k) * MSG_DIM + j], acc);
#pragma unroll
            for (int k = 0; k < EDGE_DIM; ++k)
                acc = fmaf(edge_attr[e * EDGE_DIM + k],
                           Wm[(2 * NODE_DIM + k) * MSG_DIM + j], acc);
            acc = acc > 0.0f ? acc : 0.0f;
            atomicAdd(&x[d * MSG_DIM + j], acc);
        }
    }
}

// ---------------------------------------------------------------------------
// Node phase: x -> relu(W1) -> relu(W2) -> relu(W3) -> [5], scatter to graphs.
// Memory bound (50 MFLOP over 100K nodes); scalar FMAs are the right tool.
// ---------------------------------------------------------------------------
__global__ __launch_bounds__(256) void node_mlp(
    const float* __restrict__ x,
    const int*   __restrict__ batch,
    const float* __restrict__ W1, const float* __restrict__ b1,
    const float* __restrict__ W2, const float* __restrict__ b2,
    const float* __restrict__ W3, const float* __restrict__ b3,
    float*       __restrict__ g,   // [n_graphs, 5] accumulator
    int n_nodes) {
    const int i = blockIdx.x * blockDim.x + threadIdx.x;
    if (i >= n_nodes) return;

    float v0[MSG_DIM], v1[MSG_DIM], v2[MSG_DIM], v3[HID_DIM];
#pragma unroll
    for (int j = 0; j < MSG_DIM; ++j) v0[j] = x[i * MSG_DIM + j];

#pragma unroll
    for (int j = 0; j < MSG_DIM; ++j) {
        float s = b1[j];
#pragma unroll
        for (int k = 0; k < MSG_DIM; ++k) s = fmaf(v0[k], W1[k * MSG_DIM + j], s);
        v1[j] = s > 0.0f ? s : 0.0f;
    }
#pragma unroll
    for (int j = 0; j < MSG_DIM; ++j) {
        float s = b2[j];
#pragma unroll
        for (int k = 0; k < MSG_DIM; ++k) s = fmaf(v1[k], W2[k * MSG_DIM + j], s);
        v2[j] = s > 0.0f ? s : 0.0f;
    }
#pragma unroll
    for (int j = 0; j < HID_DIM; ++j) {
        float s = b3[j];
#pragma unroll
        for (int k = 0; k < MSG_DIM; ++k) s = fmaf(v2[k], W3[k * HID_DIM + j], s);
        v3[j] = s > 0.0f ? s : 0.0f;
    }

    const int gidx = batch[i];
#pragma unroll
    for (int j = 0; j < HID_DIM; ++j)
        atomicAdd(&g[gidx * HID_DIM + j], v3[j]);
}

// ---------------------------------------------------------------------------
// Graph head: g -> relu(W4) -> W5 -> [G,1]
// ---------------------------------------------------------------------------
__global__ __launch_bounds__(256) void graph_mlp(
    const float* __restrict__ g,
    const float* __restrict__ W4, const float* __restrict__ b4,
    const float* __restrict__ W5, const float* __restrict__ b5,
    float*       __restrict__ out,
    int n_graphs) {
    const int i = blockIdx.x * blockDim.x + threadIdx.x;
    if (i >= n_graphs) return;

    float gv[HID_DIM], h[HID_DIM];
#pragma unroll
    for (int j = 0; j < HID_DIM; ++j) gv[j] = g[i * HID_DIM + j];

#pragma unroll
    for (int j = 0; j < HID_DIM; ++j) {
        float s = b4[j];
#pragma unroll
        for (int k = 0; k < HID_DIM; ++k) s = fmaf(gv[k], W4[k * HID_DIM + j], s);
        h[j] = s > 0.0f ? s : 0.0f;
    }
    float o = b5[0];
#pragma unroll
    for (int k = 0; k < HID_DIM; ++k) o = fmaf(h[k], W5[k], o);
    out[i] = o;
}

// ---------------------------------------------------------------------------
extern "C" void kernel_launch(void* const* d_in, const int* in_sizes, int n_in,
                              void* d_out, int out_size, void* d_ws, size_t ws_size,
                              hipStream_t stream) {
    const int*   edge_index = (const int*)  d_in[0];   // [2, E]
    const float* node_attr  = (const float*)d_in[1];   // [N, 8]
    const float* edge_attr  = (const float*)d_in[2];   // [E, 4]
    const int*   batch      = (const int*)  d_in[3];   // [N]
    const float* Wm = (const float*)d_in[4];
    const float* bm = (const float*)d_in[5];
    const float* W1 = (const float*)d_in[6];
    const float* b1 = (const float*)d_in[7];
    const float* W2 = (const float*)d_in[8];
    const float* b2 = (const float*)d_in[9];
    const float* W3 = (const float*)d_in[10];
    const float* b3 = (const float*)d_in[11];
    const float* W4 = (const float*)d_in[12];
    const float* b4 = (const float*)d_in[13];
    const float* W5 = (const float*)d_in[14];
    const float* b5 = (const float*)d_in[15];

    const int n_edges  = in_sizes[0] / 2;
    const int n_nodes  = in_sizes[1] / NODE_DIM;
    const int n_graphs = out_size;                     // output is [G, 1]

    const int* src = edge_index;                       // row 0
    const int* dst = edge_index + n_edges;             // row 1

    float* x_acc = (float*)d_ws;                       // [N, 10]
    float* g_acc = x_acc + (size_t)n_nodes * MSG_DIM;  // [G, 5]
    const int zero_n = n_nodes * MSG_DIM + n_graphs * HID_DIM;

    // 1) zero accumulators
    {
        int blocks = (zero_n + 255) / 256;
        if (blocks > 2048) blocks = 2048;
        zero_ws<<<blocks, 256, 0, stream>>>(x_acc, zero_n);
    }
    // 2) edge phase (WMMA): 16 edges/wave-tile, grid-stride, 8 waves/block.
    //    Fixed grid so B fragments are loaded once and reused ~12x per wave.
    {
        const int tiles = (n_edges + 15) / 16;
        int blocks = (tiles + 7) / 8;
        if (blocks > 2048) blocks = 2048;              // ~16K waves fills the GPU
        edge_mlp_wmma<<<blocks, 256, 0, stream>>>(
            src, dst, node_attr, edge_attr, Wm, bm, x_acc, n_edges);
    }
    // 3) node phase
    {
        const int blocks = (n_nodes + 255) / 256;
        node_mlp<<<blocks, 256, 0, stream>>>(
            x_acc, batch, W1, b1, W2, b2, W3, b3, g_acc, n_nodes);
    }
    // 4) graph head
    {
        const int blocks = (n_graphs + 255) / 256;
        graph_mlp<<<blocks, 256, 0, stream>>>(
            g_acc, W4, b4, W5, b5, (float*)d_out, n_graphs);
    }
}